// CVAE_29497835389865
// MI455X (gfx1250) — compile-verified
//
#include <hip/hip_runtime.h>
#include <math.h>

// ---------------------------------------------------------------------------
// CVAE ladder, single persistent workgroup, all matvecs via V_WMMA_F32_16X16X4_F32
// Branchless result extraction; weights, biases, eps all staged in LDS.
// ---------------------------------------------------------------------------

typedef __attribute__((ext_vector_type(2))) float v2f;
typedef __attribute__((ext_vector_type(8))) float v8f;

#define NT 256
#define NW 8

// LDS layout (floats). Weights zero-padded: M -> mult of 16, K -> mult of 4.
constexpr int O_W1  = 0;                 // 112 x 40  (100 x 38)
constexpr int O_W2M = O_W1  + 112 * 40;  // 32 x 100  (28 x 100)
constexpr int O_W2S = O_W2M + 32 * 100;  // 32 x 100
constexpr int O_W3  = O_W2S + 32 * 100;  // 112 x 40  (100 x 38)
constexpr int O_W4  = O_W3  + 112 * 40;  // 32 x 100
constexpr int O_W5  = O_W4  + 32 * 100;  // 112 x 28  (100 x 28)
constexpr int O_W6  = O_W5  + 112 * 28;  // 32 x 100
// zero-padded biases
constexpr int O_B1  = O_W6  + 32 * 100;  // 112
constexpr int O_B2M = O_B1  + 112;       // 32
constexpr int O_B2S = O_B2M + 32;        // 32
constexpr int O_B3  = O_B2S + 32;        // 112
constexpr int O_B4  = O_B3  + 112;       // 32
constexpr int O_B5  = O_B4  + 32;        // 112
constexpr int O_B6  = O_B5  + 112;       // 32
constexpr int O_EPS = O_B6  + 32;        // 112 (4 x 28)
// vectors
constexpr int O_VIN = O_EPS + 112;       // 48   : [v(28); y(10); zero pad]
constexpr int O_H   = O_VIN + 48;        // 112  : hidden (100 valid)
constexpr int O_MU  = O_H   + 112;       // 32
constexpr int O_LS  = O_MU  + 32;        // 32
constexpr int O_R   = O_LS  + 32;        // 48   : recon (28 valid)
constexpr int LDS_F = O_R + 48;
constexpr int LDS_BYTES = LDS_F * (int)sizeof(float);

enum { ACT_NONE = 0, ACT_RELU = 1, ACT_SIG = 2 };

__device__ __forceinline__ float quant1(float z) {
  // nearest of 9 codebook values; first-min on ties (matches jnp.argmin)
  float best = 0.25f, bd = fabsf(z - 0.25f);
  const float cbs[8] = {0.3536f, 0.5f, 0.7071f, 1.0f, 1.4142f, 2.0f, 2.8284f, 4.0f};
#pragma unroll
  for (int i = 0; i < 8; ++i) {
    float d = fabsf(z - cbs[i]);
    if (d < bd) { bd = d; best = cbs[i]; }
  }
  return best;
}

// out[0..Mp) = act( W(Mp x Kp, LDS, padded) @ x(Kp, LDS) + biasLds(Mp, padded) )
// A frag: lanes 0-15 hold K{0,1} in v0/v1; lanes 16-31 hold K{2,3}. (ISA 7.12.2)
// B frag: vector replicated across all 16 N columns -> every column of D is W@x,
//         so EVERY lane holds the full 8-row result for its half.
// Extraction is fully branchless: lane L picks c[L&7] (cndmask chain), writes
// row m0 + 8*(L>=16) + (L&7). Duplicate lanes store identical values.
template <int ACT>
__device__ __forceinline__ void matvec_wmma(
    const float* Wl, int Mp, int Kp,
    const float* xl, float* outl, const float* bl,
    int lane, int wave, int nwaves)
{
  const int half = lane >> 4;   // 0: K pair {0,1}, 1: K pair {2,3}
  const int lrow = lane & 15;
  const int idx  = lane & 7;
  const int Mtiles = Mp >> 4;
  const int Ktiles = Kp >> 2;
  for (int t = wave; t < Mtiles; t += nwaves) {   // wave-uniform: EXEC all-ones
    const int m0 = t << 4;
    v8f c = {0.f, 0.f, 0.f, 0.f, 0.f, 0.f, 0.f, 0.f};
    const float* wbase = Wl + (m0 + lrow) * Kp + (half << 1);
    for (int k = 0; k < Ktiles; ++k) {
      v2f a, b;
      a.x = wbase[4 * k + 0];
      a.y = wbase[4 * k + 1];
      b.x = xl[4 * k + (half << 1) + 0];
      b.y = xl[4 * k + (half << 1) + 1];
      c = __builtin_amdgcn_wmma_f32_16x16x4_f32(false, a, false, b,
                                                (short)0, c, false, false);
    }
    // branchless select of c[idx]
    float v = c[0];
#pragma unroll
    for (int r = 1; r < 8; ++r) v = (idx == r) ? c[r] : v;
    const int row = m0 + (half << 3) + idx;
    v += bl[row];
    if (ACT == ACT_RELU) v = fmaxf(v, 0.f);
    if (ACT == ACT_SIG)  v = 1.f / (1.f + __expf(-v));
    outl[row] = v;
  }
}

__global__ void __launch_bounds__(NT)
cvae_pipeline(const float* __restrict__ x, const float* __restrict__ y,
              const float* __restrict__ eps,
              const float* __restrict__ W1,  const float* __restrict__ b1,
              const float* __restrict__ W2m, const float* __restrict__ b2m,
              const float* __restrict__ W2s, const float* __restrict__ b2s,
              const float* __restrict__ W3,  const float* __restrict__ b3,
              const float* __restrict__ W4,  const float* __restrict__ b4,
              const float* __restrict__ W5,  const float* __restrict__ b5,
              const float* __restrict__ W6,  const float* __restrict__ b6,
              float* __restrict__ out)
{
  extern __shared__ float lds[];
  const int tid  = threadIdx.x;
  const int lane = tid & 31, wave = tid >> 5;

  // ---- stage weights + biases + eps into LDS once (zero-padded) ----
  auto loadW = [&](const float* g, int M, int K, int Mp, int Kp, int off) {
    const int n = Mp * Kp;
    for (int i = tid; i < n; i += NT) {
      const int m = i / Kp, k = i - m * Kp;
      lds[off + i] = (m < M && k < K) ? g[m * K + k] : 0.f;
    }
  };
  auto loadB = [&](const float* g, int M, int Mp, int off) {
    for (int i = tid; i < Mp; i += NT) lds[off + i] = (i < M) ? g[i] : 0.f;
  };
  loadW(W1,  100, 38, 112, 40,  O_W1);
  loadW(W2m, 28, 100, 32,  100, O_W2M);
  loadW(W2s, 28, 100, 32,  100, O_W2S);
  loadW(W3,  100, 38, 112, 40,  O_W3);
  loadW(W4,  28, 100, 32,  100, O_W4);
  loadW(W5,  100, 28, 112, 28,  O_W5);
  loadW(W6,  28, 100, 32,  100, O_W6);
  loadB(b1, 100, 112, O_B1);
  loadB(b2m, 28, 32,  O_B2M);
  loadB(b2s, 28, 32,  O_B2S);
  loadB(b3, 100, 112, O_B3);
  loadB(b4,  28, 32,  O_B4);
  loadB(b5, 100, 112, O_B5);
  loadB(b6,  28, 32,  O_B6);
  if (tid < 112) lds[O_EPS + tid] = eps[tid];
  for (int i = tid; i < LDS_F - O_VIN; i += NT) lds[O_VIN + i] = 0.f;
  if (tid < 28) { out[112 + tid] = 0.f; out[336 + tid] = 0.f; } // mu_e row3, logstd row3
  __syncthreads();
  if (tid < 28)      lds[O_VIN + tid] = x[tid];
  else if (tid < 38) lds[O_VIN + tid] = y[tid - 28];
  __syncthreads();

  // ---- 4 encoder levels: vin=[v;y] -> h -> (mu, sigmoid(logstd)) -> quantize ----
  for (int lvl = 0; lvl < 4; ++lvl) {
    matvec_wmma<ACT_RELU>(lds + O_W1, 112, 40, lds + O_VIN, lds + O_H,
                          lds + O_B1, lane, wave, NW);
    __syncthreads();
    if (wave < 4)   // wave-uniform split: mu on waves 0-3, logstd on waves 4-7
      matvec_wmma<ACT_NONE>(lds + O_W2M, 32, 100, lds + O_H, lds + O_MU,
                            lds + O_B2M, lane, wave, 4);
    else
      matvec_wmma<ACT_SIG>(lds + O_W2S, 32, 100, lds + O_H, lds + O_LS,
                           lds + O_B2S, lane, wave - 4, 4);
    __syncthreads();
    if (tid < 28) {
      const float mu = lds[O_MU + tid], ls = lds[O_LS + tid];
      if (lvl < 3) {
        out[28  + lvl * 28 + tid] = mu;   // mu_e rows 0..2
        out[252 + lvl * 28 + tid] = ls;   // logstd rows 0..2
      }
      lds[O_VIN + tid] = quant1(mu + lds[O_EPS + lvl * 28 + tid] * ls);
    }
    __syncthreads();
  }

  // ---- 4 decoder levels, i = 3..0:
  //  recon_i = sigmoid(W4 @ relu(W3 @ [z;y] + b3) + b4)
  //  mud_i   = W6 @ (W5 @ recon_i + b5) + b6 ;  next z = quant(recon_i) ----
  for (int i = 3; i >= 0; --i) {
    matvec_wmma<ACT_RELU>(lds + O_W3, 112, 40, lds + O_VIN, lds + O_H,
                          lds + O_B3, lane, wave, NW);
    __syncthreads();
    matvec_wmma<ACT_SIG>(lds + O_W4, 32, 100, lds + O_H, lds + O_R,
                         lds + O_B4, lane, wave, NW);
    __syncthreads();
    matvec_wmma<ACT_NONE>(lds + O_W5, 112, 28, lds + O_R, lds + O_H,
                          lds + O_B5, lane, wave, NW);
    __syncthreads();
    matvec_wmma<ACT_NONE>(lds + O_W6, 32, 100, lds + O_H, lds + O_MU,
                          lds + O_B6, lane, wave, NW);
    __syncthreads();
    if (tid < 28) {
      out[140 + i * 28 + tid] = lds[O_MU + tid];          // mu_d row i
      if (i > 0) lds[O_VIN + tid] = quant1(lds[O_R + tid]);
      else       out[tid] = lds[O_R + tid];               // recon0
    }
    __syncthreads();
  }
}

extern "C" void kernel_launch(void* const* d_in, const int* in_sizes, int n_in,
                              void* d_out, int out_size, void* d_ws, size_t ws_size,
                              hipStream_t stream) {
  (void)in_sizes; (void)n_in; (void)out_size; (void)d_ws; (void)ws_size;
  const float* x   = (const float*)d_in[0];
  const float* yv  = (const float*)d_in[1];
  const float* eps = (const float*)d_in[2];
  const float* W1  = (const float*)d_in[3];  const float* b1  = (const float*)d_in[4];
  const float* W2m = (const float*)d_in[5];  const float* b2m = (const float*)d_in[6];
  const float* W2s = (const float*)d_in[7];  const float* b2s = (const float*)d_in[8];
  const float* W3  = (const float*)d_in[9];  const float* b3  = (const float*)d_in[10];
  const float* W4  = (const float*)d_in[11]; const float* b4  = (const float*)d_in[12];
  const float* W5  = (const float*)d_in[13]; const float* b5  = (const float*)d_in[14];
  const float* W6  = (const float*)d_in[15]; const float* b6  = (const float*)d_in[16];

  // ~100 KB dynamic LDS (WGP supports up to 320 KB) — raise the per-kernel cap.
  hipFuncSetAttribute(reinterpret_cast<const void*>(cvae_pipeline),
                      hipFuncAttributeMaxDynamicSharedMemorySize, LDS_BYTES);

  cvae_pipeline<<<1, NT, LDS_BYTES, stream>>>(
      x, yv, eps, W1, b1, W2m, b2m, W2s, b2s, W3, b3, W4, b4, W5, b5, W6, b6,
      (float*)d_out);
}